// QuantumNet_77627238908011
// MI455X (gfx1250) — compile-verified
//
#include <hip/hip_runtime.h>
#include <math.h>

#define NQ     12
#define NL     3
#define NSTATE 4096   // 2^12
#define NSAMP  2048
#define TS     64     // output tile (64x64) per workgroup
#define KC     32     // k-chunk staged in LDS
#define NCH    (NSTATE / KC)              // 128 chunks
#define LDA    36     // padded LDS row stride (floats): 144B rows -> 16B aligned
#define BUFB   (TS * LDA * 4)             // bytes per LDS panel buffer

typedef float v2f __attribute__((ext_vector_type(2)));
typedef float v8f __attribute__((ext_vector_type(8)));

// ---------------------------------------------------------------------------
// Kernel 1: statevector simulation, one sample per workgroup, state in LDS.
// Reference flattens row-major with qubit 0 as MSB -> qubit q lives at bit 11-q.
// ---------------------------------------------------------------------------
__global__ __launch_bounds__(256)
void sim_kernel(const float* __restrict__ data,    // (2048,12)
                const float* __restrict__ params,  // (3,12)
                float* __restrict__ psi)           // (2048,4096)
{
  __shared__ float st[NSTATE];
  const int tid  = threadIdx.x;
  const int samp = blockIdx.x;

  for (int i = tid; i < NSTATE; i += 256) st[i] = 0.f;
  if (tid == 0) st[0] = 1.f;
  __syncthreads();

  for (int l = 0; l < NL; ++l) {
    for (int q = 0; q < NQ; ++q) {
      const float theta = params[l * NQ + q] + data[samp * NQ + q];
      float s, c;
      __sincosf(0.5f * theta, &s, &c);
      const int m = 1 << (NQ - 1 - q);
      for (int p = tid; p < NSTATE / 2; p += 256) {
        const int low  = p & (m - 1);
        const int idx0 = ((p ^ low) << 1) | low;  // insert 0 at bit of qubit q
        const int idx1 = idx0 | m;
        const float a0 = st[idx0];
        const float a1 = st[idx1];
        st[idx0] = c * a0 - s * a1;
        st[idx1] = s * a0 + c * a1;
      }
      __syncthreads();
    }
    // CZ ring: flip sign when an adjacent (ring) pair of qubit bits are both 1.
    for (int i = tid; i < NSTATE; i += 256) {
      const unsigned b   = (unsigned)i;
      const unsigned adj = b & (((b >> 1) | (b << 11)) & 0xFFFu);
      if (__popc(adj) & 1) st[i] = -st[i];
    }
    __syncthreads();
  }

  for (int i = tid; i < NSTATE; i += 256)
    psi[(size_t)samp * NSTATE + i] = st[i];
}

// ---------------------------------------------------------------------------
// Kernel 2: fused Gram + reduction via fp32 WMMA (V_WMMA_F32_16X16X4_F32).
// 64x64 output tile per 256-thread block; 8 waves x two 16x16 subtiles.
// Global->LDS panels move via async DMA (GLOBAL_LOAD_ASYNC_TO_LDS_B128,
// ASYNCcnt-tracked) double-buffered so chunk c+1's DMA overlaps chunk c's
// WMMAs. Never materializes K: accumulates s1 = sum l_i l_j G^2, s2 = sum G^4.
// ---------------------------------------------------------------------------
__global__ __launch_bounds__(256)
void gram_kernel(const float* __restrict__ psi,
                 const int*   __restrict__ labels,
                 float* __restrict__ partial1,
                 float* __restrict__ partial2)
{
  __shared__ float As[2][TS * LDA];
  __shared__ float Bs[2][TS * LDA];
  __shared__ float labI[TS];
  __shared__ float labJ[TS];
  __shared__ float red1[8];
  __shared__ float red2[8];

  const int tid  = threadIdx.x;
  const int lane = tid & 31;
  const int wave = tid >> 5;
  const int i0   = blockIdx.y * TS;
  const int j0   = blockIdx.x * TS;

  if (tid < TS)            labI[tid]      = 2.f * (float)labels[i0 + tid] - 1.f;
  else if (tid < 2 * TS)   labJ[tid - TS] = 2.f * (float)labels[j0 + (tid - TS)] - 1.f;

  const int mi  = wave & 3;         // row subtile 0..3
  const int nj0 = (wave >> 2) * 2;  // col subtiles nj0, nj0+1
  const int hf  = lane >> 4;        // lane half (K slot selector)
  const int m16 = lane & 15;

  // cooperative DMA coords: each thread moves 2 rows x 16B per panel per chunk
  const int lrow = tid >> 3;        // 0..31
  const int lcol = (tid & 7) * 4;   // 0..28 (floats)

  // global source addresses (bytes); +kc*4 per chunk
  const unsigned long long gA0 =
      (unsigned long long)(uintptr_t)&psi[(size_t)(i0 + lrow) * NSTATE + lcol];
  const unsigned long long gA1 = gA0 + (unsigned long long)32 * NSTATE * 4;
  const unsigned long long gB0 =
      (unsigned long long)(uintptr_t)&psi[(size_t)(j0 + lrow) * NSTATE + lcol];
  const unsigned long long gB1 = gB0 + (unsigned long long)32 * NSTATE * 4;

  // LDS destination offsets: low 32 bits of a flat shared pointer ARE the LDS
  // byte offset (flat LDS aperture).
  const unsigned ldsA0 = (unsigned)(uintptr_t)&As[0][lrow * LDA + lcol];
  const unsigned ldsA1 = (unsigned)(uintptr_t)&As[0][(lrow + 32) * LDA + lcol];
  const unsigned ldsB0 = (unsigned)(uintptr_t)&Bs[0][lrow * LDA + lcol];
  const unsigned ldsB1 = (unsigned)(uintptr_t)&Bs[0][(lrow + 32) * LDA + lcol];

  auto issue = [&](int buf, int kc) {
    const unsigned bo = (unsigned)(buf * BUFB);
    const unsigned long long ko = (unsigned long long)kc * 4ull;
    asm volatile("global_load_async_to_lds_b128 %0, %1, off"
                 :: "v"(ldsA0 + bo), "v"(gA0 + ko) : "memory");
    asm volatile("global_load_async_to_lds_b128 %0, %1, off"
                 :: "v"(ldsA1 + bo), "v"(gA1 + ko) : "memory");
    asm volatile("global_load_async_to_lds_b128 %0, %1, off"
                 :: "v"(ldsB0 + bo), "v"(gB0 + ko) : "memory");
    asm volatile("global_load_async_to_lds_b128 %0, %1, off"
                 :: "v"(ldsB1 + bo), "v"(gB1 + ko) : "memory");
  };

  v8f c0 = {};
  v8f c1 = {};

  issue(0, 0);  // prologue: DMA chunk 0 into buffer 0

  for (int c = 0; c < NCH; ++c) {
    const int cur = c & 1;
    // my DMA into buffer `cur` has landed in LDS
    asm volatile("s_wait_asynccnt 0x0" ::: "memory");
    // everyone's DMA landed AND everyone consumed buffer cur^1 (fragment
    // ds_loads are DScnt-waited before the WMMAs preceding this barrier)
    __syncthreads();
    if (c + 1 < NCH) issue(cur ^ 1, (c + 1) * KC);  // overlap next chunk's DMA

    const float* __restrict__ Ac = &As[cur][0];
    const float* __restrict__ Bc = &Bs[cur][0];
#pragma unroll
    for (int kk = 0; kk < KC; kk += 4) {
      const int ko = kk + 2 * hf;   // this lane-half's K pair (8B aligned)
      const v2f a  = *(const v2f*)&Ac[(mi * 16 + m16) * LDA + ko];
      const v2f b0 = *(const v2f*)&Bc[(nj0 * 16 + m16) * LDA + ko];
      const v2f b1 = *(const v2f*)&Bc[((nj0 + 1) * 16 + m16) * LDA + ko];
      c0 = __builtin_amdgcn_wmma_f32_16x16x4_f32(false, a, false, b0, (short)0, c0, false, false);
      c1 = __builtin_amdgcn_wmma_f32_16x16x4_f32(false, a, false, b1, (short)0, c1, false, false);
    }
  }

  // C/D layout: VGPR v holds row M = v + 8*(lane>>4), col N = lane&15
  float s1 = 0.f, s2 = 0.f;
  const float lj0v = labJ[nj0 * 16 + m16];
  const float lj1v = labJ[(nj0 + 1) * 16 + m16];
#pragma unroll
  for (int v = 0; v < 8; ++v) {
    const float li = labI[mi * 16 + v + 8 * hf];
    const float g0 = c0[v];
    const float k0 = g0 * g0;
    s1 += li * lj0v * k0;
    s2 += k0 * k0;
    const float g1 = c1[v];
    const float k1 = g1 * g1;
    s1 += li * lj1v * k1;
    s2 += k1 * k1;
  }

  // wave32 reduction, then block reduction in fixed order (deterministic)
  for (int off = 16; off > 0; off >>= 1) {
    s1 += __shfl_down(s1, off, 32);
    s2 += __shfl_down(s2, off, 32);
  }
  if (lane == 0) { red1[wave] = s1; red2[wave] = s2; }
  __syncthreads();
  if (tid == 0) {
    float t1 = 0.f, t2 = 0.f;
    for (int w = 0; w < 8; ++w) { t1 += red1[w]; t2 += red2[w]; }
    const int bid = blockIdx.y * gridDim.x + blockIdx.x;
    partial1[bid] = t1;
    partial2[bid] = t2;
  }
}

// ---------------------------------------------------------------------------
// Kernel 3: deterministic final reduction. sum(LL^2) = N^2 = 2048^2.
// out = s1 / sqrt(s2 * 2048^2) = s1 / (2048 * sqrt(s2))
// ---------------------------------------------------------------------------
__global__ void finalize_kernel(const float* __restrict__ p1,
                                const float* __restrict__ p2,
                                int n, float* __restrict__ out)
{
  if (threadIdx.x == 0 && blockIdx.x == 0) {
    float s1 = 0.f, s2 = 0.f;
    for (int i = 0; i < n; ++i) { s1 += p1[i]; s2 += p2[i]; }
    out[0] = s1 / (2048.f * sqrtf(s2));
  }
}

extern "C" void kernel_launch(void* const* d_in, const int* in_sizes, int n_in,
                              void* d_out, int out_size, void* d_ws, size_t ws_size,
                              hipStream_t stream) {
  const float* data   = (const float*)d_in[0];  // (2048,12) f32
  const int*   labels = (const int*)  d_in[1];  // (2048,)   i32
  const float* params = (const float*)d_in[2];  // (3,12)    f32
  float* out = (float*)d_out;

  float* psi = (float*)d_ws;                               // 2048*4096 f32 = 32MB
  const int nTiles = (NSAMP / TS) * (NSAMP / TS);          // 32*32 = 1024
  float* p1 = psi + (size_t)NSAMP * NSTATE;
  float* p2 = p1 + nTiles;

  sim_kernel<<<NSAMP, 256, 0, stream>>>(data, params, psi);

  dim3 grid(NSAMP / TS, NSAMP / TS);
  gram_kernel<<<grid, 256, 0, stream>>>(psi, labels, p1, p2);

  finalize_kernel<<<1, 64, 0, stream>>>(p1, p2, nTiles, out);
}